// ScalarCounterPDA_13022340842134
// MI455X (gfx1250) — compile-verified
//
#include <hip/hip_runtime.h>

// ---------------------------------------------------------------------------
// ScalarCounterPDA on MI455X (gfx1250, wave32)
//
// Phase 1 (k_reduce): per-block composite of the clamped-add monoid
// Phase 2 (k_spine):  serial scan over 128 block composites (trivial)
// Phase 3 (k_emit):   per-block exact replay + WMMA-tiled softmax readout
//
// The problem is store-bandwidth bound (100 MB out @ 23.3 TB/s ~ 4.5 us),
// so compute per element is kept negligible and stores are coalesced +
// non-temporal (write-once stream).
// ---------------------------------------------------------------------------

#define T_LEN   (1 << 20)
#define BLOCK   256
#define PT      32                      // elements per thread
#define CHUNK   (BLOCK * PT)            // 8192 elements per block
#define NBLK    (T_LEN / CHUNK)         // 128 blocks
#define NWAVES  (BLOCK / 32)            // 8 waves (wave32)
#define NSYM    16
#define NOUT    25
#define CMAX    24.0f

typedef float v2f __attribute__((ext_vector_type(2)));
typedef float v8f __attribute__((ext_vector_type(8)));

// clamped-add transfer function f(c) = clip(c + b, lo, hi); closed under
// composition, which makes the "non-associative" scan parallelizable.
struct Tri { float b, lo, hi; };

__device__ __forceinline__ Tri tri_identity() {
  Tri t; t.b = 0.0f; t.lo = -1e30f; t.hi = 1e30f; return t;
}

// apply f first, then g
__device__ __forceinline__ Tri tri_combine(Tri f, Tri g) {
  Tri h;
  h.b  = f.b + g.b;
  h.lo = fminf(fmaxf(f.lo + g.b, g.lo), g.hi);
  h.hi = fminf(fmaxf(f.hi + g.b, g.lo), g.hi);
  return h;
}

__device__ __forceinline__ float tri_apply(Tri f, float c) {
  return fminf(fmaxf(c + f.b, f.lo), f.hi);
}

// compose one recurrence step clip(x + d, 0, CMAX) onto f (f first)
__device__ __forceinline__ Tri tri_step(Tri f, float d) {
  Tri h;
  h.b  = f.b + d;
  h.lo = fminf(fmaxf(f.lo + d, 0.0f), CMAX);
  h.hi = fminf(fmaxf(f.hi + d, 0.0f), CMAX);
  return h;
}

// ---------------------------------------------------------------------------
// Phase 1: each block emits the composite transfer function of its chunk.
// ---------------------------------------------------------------------------
__global__ void __launch_bounds__(BLOCK)
k_reduce(const int* __restrict__ seq, const float* __restrict__ delta,
         float* __restrict__ tri_out) {
  __shared__ int   s_seq[CHUNK];
  __shared__ float s_delta[NSYM];
  __shared__ Tri   s_wave[NWAVES];

  const int tid = threadIdx.x;
  if (tid < NSYM) s_delta[tid] = delta[tid];

  // 128-bit coalesced staging of the input chunk into LDS
  const int4* g4 = (const int4*)(seq + (size_t)blockIdx.x * CHUNK);
  int4* l4 = (int4*)s_seq;
  for (int i = tid; i < CHUNK / 4; i += BLOCK) l4[i] = g4[i];
  __syncthreads();

  // serial composite over this thread's 32 contiguous elements
  Tri t = tri_identity();
  const int base = tid * PT;
  #pragma unroll
  for (int k = 0; k < PT; ++k)
    t = tri_step(t, s_delta[s_seq[base + k] & (NSYM - 1)]);

  // ordered wave32 tree reduce: lane i accumulates segments [i .. i+2*off-1]
  #pragma unroll
  for (int off = 1; off < 32; off <<= 1) {
    Tri o;
    o.b  = __shfl_down(t.b,  off);
    o.lo = __shfl_down(t.lo, off);
    o.hi = __shfl_down(t.hi, off);
    t = tri_combine(t, o);            // earlier(t) then later(o)
  }
  if ((tid & 31) == 0) s_wave[tid >> 5] = t;
  __syncthreads();

  if (tid == 0) {
    Tri a = s_wave[0];
    for (int w = 1; w < NWAVES; ++w) a = tri_combine(a, s_wave[w]);
    tri_out[3 * blockIdx.x + 0] = a.b;
    tri_out[3 * blockIdx.x + 1] = a.lo;
    tri_out[3 * blockIdx.x + 2] = a.hi;
  }
}

// ---------------------------------------------------------------------------
// Phase 2: serial spine over 128 block composites -> per-block start counter.
// ---------------------------------------------------------------------------
__global__ void k_spine(const float* __restrict__ tri_in,
                        float* __restrict__ starts) {
  if (threadIdx.x == 0 && blockIdx.x == 0) {
    float c = 0.0f;
    for (int i = 0; i < NBLK; ++i) {
      starts[i] = c;
      Tri f;
      f.b  = tri_in[3 * i + 0];
      f.lo = tri_in[3 * i + 1];
      f.hi = tri_in[3 * i + 2];
      c = tri_apply(f, c);
    }
  }
}

// ---------------------------------------------------------------------------
// Phase 3: exact counter replay + softmax readout.
// Readout uses V_WMMA_F32_16X16X4_F32 to build D[t][j] = pos_j - c_t for a
// 16x16 tile in one chained pair of WMMAs. A/B values are K-invariant so the
// result does not depend on the K<->(VGPR, lane-half) mapping; only the
// documented m = lane%16 (A) and n = lane%16 (B/C/D) mappings matter.
// ---------------------------------------------------------------------------
__global__ void __launch_bounds__(BLOCK)
k_emit(const int* __restrict__ seq, const float* __restrict__ delta,
       const float* __restrict__ bias, const float* __restrict__ scale_p,
       const float* __restrict__ starts, float* __restrict__ out) {
  __shared__ int   s_buf[CHUNK];      // symbols, then counters (aliased)
  __shared__ float s_delta[NSYM];
  __shared__ Tri   s_wave[NWAVES];

  const int tid  = threadIdx.x;
  const int lane = tid & 31;
  const int wid  = tid >> 5;

  if (tid < NSYM) s_delta[tid] = delta[tid];

  const int4* g4 = (const int4*)(seq + (size_t)blockIdx.x * CHUNK);
  int4* l4 = (int4*)s_buf;
  for (int i = tid; i < CHUNK / 4; i += BLOCK) l4[i] = g4[i];
  __syncthreads();

  // per-thread composite
  Tri t = tri_identity();
  const int base = tid * PT;
  #pragma unroll
  for (int k = 0; k < PT; ++k)
    t = tri_step(t, s_delta[s_buf[base + k] & (NSYM - 1)]);

  // wave32 inclusive scan of transfer functions
  #pragma unroll
  for (int off = 1; off < 32; off <<= 1) {
    Tri o;
    o.b  = __shfl_up(t.b,  off);
    o.lo = __shfl_up(t.lo, off);
    o.hi = __shfl_up(t.hi, off);
    if (lane >= off) t = tri_combine(o, t);   // earlier(o) then current(t)
  }
  Tri excl;
  excl.b  = __shfl_up(t.b,  1);
  excl.lo = __shfl_up(t.lo, 1);
  excl.hi = __shfl_up(t.hi, 1);
  if (lane == 0) excl = tri_identity();
  if (lane == 31) s_wave[wid] = t;
  __syncthreads();

  Tri pref = tri_identity();
  for (int w = 0; w < wid; ++w) pref = tri_combine(pref, s_wave[w]);
  Tri myexcl = tri_combine(pref, excl);
  float c = tri_apply(myexcl, starts[blockIdx.x]);

  // exact replay; overwrite symbol slot with the f32 counter (same owner)
  #pragma unroll
  for (int k = 0; k < PT; ++k) {
    const float d = s_delta[s_buf[base + k] & (NSYM - 1)];
    c = fminf(fmaxf(c + d, 0.0f), CMAX);
    s_buf[base + k] = __float_as_int(c);
  }
  __syncthreads();

  // -------- WMMA-tiled softmax readout: 16 rows (t's) per wave-iteration ----
  const int   col  = lane & 15;         // n for tile0, n-16 for tile1
  const int   half = lane >> 4;         // C/D rows split: m = r + 8*half
  const float scale = scale_p[0];
  const float bn0 = bias[col];
  const int   n1  = 16 + col;
  const bool  v1  = (n1 < NOUT);
  const float bn1 = v1 ? bias[n1] : 0.0f;

  const v2f aq   = {0.25f, 0.25f};                       // A = 1/4 (all k)
  const v2f bneg = {-0.25f, -0.25f};                     // B = -1/4 (all k)
  const v2f bp0  = {(float)col, (float)col};             // B = pos_n, tile0
  const v2f bp1  = {(float)n1,  (float)n1};              // B = pos_n, tile1
  const int rowBlock = blockIdx.x * CHUNK;

  for (int g = wid; g < CHUNK / 16; g += NWAVES) {        // uniform per wave
    const int r0 = g * 16;
    const float cm = __int_as_float(s_buf[r0 + col]);     // c of row m=lane%16
    v2f ac = {cm, cm};                                    // A = c_m (all k)
    v8f z  = {0.f, 0.f, 0.f, 0.f, 0.f, 0.f, 0.f, 0.f};

    // acc[m][n] = sum_k c_m * (-1/4) = -c_m   (broadcast over n)
    v8f acc = __builtin_amdgcn_wmma_f32_16x16x4_f32(
        false, ac, false, bneg, (short)0, z, false, false);
    // s[m][n] = sum_k (1/4) * pos_n + acc = pos_n - c_m
    v8f s0 = __builtin_amdgcn_wmma_f32_16x16x4_f32(
        false, aq, false, bp0, (short)0, acc, false, false);
    v8f s1 = __builtin_amdgcn_wmma_f32_16x16x4_f32(
        false, aq, false, bp1, (short)0, acc, false, false);

    // numerators for all 8 rows first
    float p0v[8], p1v[8], q[8];
    #pragma unroll
    for (int r = 0; r < 8; ++r) {
      p0v[r] = __expf(bn0 - scale * fabsf(s0[r]));
      p1v[r] = v1 ? __expf(bn1 - scale * fabsf(s1[r])) : 0.0f;
      q[r]   = p0v[r] + p1v[r];
    }
    // softmax denominators: batched butterfly over the 16 lanes of each half.
    // Batching all 8 rows per level lets the 8 ds_bpermutes issue before a
    // single in-order dscnt wait instead of 32 dependent round-trips.
    #pragma unroll
    for (int mask = 1; mask <= 8; mask <<= 1) {
      float tq[8];
      #pragma unroll
      for (int r = 0; r < 8; ++r) tq[r] = __shfl_xor(q[r], mask);
      #pragma unroll
      for (int r = 0; r < 8; ++r) q[r] += tq[r];
    }
    #pragma unroll
    for (int r = 0; r < 8; ++r) {
      const float inv = __builtin_amdgcn_rcpf(q[r]);      // v_rcp_f32, ~1 ulp
      const int row = rowBlock + r0 + r + (half << 3);
      float* o = out + (size_t)row * NOUT;
      __builtin_nontemporal_store(p0v[r] * inv, o + col); // 64B burst per half
      if (v1) __builtin_nontemporal_store(p1v[r] * inv, o + n1);
    }
  }
}

// ---------------------------------------------------------------------------
extern "C" void kernel_launch(void* const* d_in, const int* in_sizes, int n_in,
                              void* d_out, int out_size, void* d_ws,
                              size_t ws_size, hipStream_t stream) {
  (void)in_sizes; (void)n_in; (void)out_size; (void)ws_size;
  const int*   seq   = (const int*)d_in[0];
  const float* delta = (const float*)d_in[1];
  const float* bias  = (const float*)d_in[2];
  const float* scale = (const float*)d_in[3];
  float* out = (float*)d_out;

  float* tri    = (float*)d_ws;        // 3*NBLK floats
  float* starts = tri + 3 * NBLK;      // NBLK floats

  k_reduce<<<NBLK, BLOCK, 0, stream>>>(seq, delta, tri);
  k_spine<<<1, 32, 0, stream>>>(tri, starts);
  k_emit<<<NBLK, BLOCK, 0, stream>>>(seq, delta, bias, scale, starts, out);
}